// NeuralNetwork_48893907698177
// MI455X (gfx1250) — compile-verified
//
#include <hip/hip_runtime.h>

typedef float v2f __attribute__((ext_vector_type(2)));
typedef float v8f __attribute__((ext_vector_type(8)));

#define DIM 256  // embedding dim (fixed by reference)

// ---------------------------------------------------------------------------
// CDNA5 async global->LDS helpers (ASYNCcnt-tracked, see CDNA5 ISA)
// ---------------------------------------------------------------------------
__device__ __forceinline__ void async_load_lds_b128(unsigned ldsAddr, const void* gaddr) {
    asm volatile("global_load_async_to_lds_b128 %0, %1, off"
                 :: "v"(ldsAddr), "v"(gaddr) : "memory");
}
__device__ __forceinline__ void wait_asynccnt0() {
#if __has_builtin(__builtin_amdgcn_s_wait_asynccnt)
    __builtin_amdgcn_s_wait_asynccnt(0);
#else
    asm volatile("s_wait_asynccnt 0x0" ::: "memory");
#endif
}

// ---------------------------------------------------------------------------
// init: zero the loss accumulator
// ---------------------------------------------------------------------------
__global__ void vq_init_kernel(float* __restrict__ lossAcc) {
    if (threadIdx.x == 0) lossAcc[0] = 0.0f;
}

// ---------------------------------------------------------------------------
// codebook row norms: ||E_k||^2, one wave (32 lanes) per row
// ---------------------------------------------------------------------------
__global__ __launch_bounds__(256) void vq_norms_kernel(
    const float* __restrict__ cb, float* __restrict__ norms, int K) {
    int row  = blockIdx.x * 8 + (threadIdx.x >> 5);
    int lane = threadIdx.x & 31;
    if (row >= K) return;
    const float* r = cb + (size_t)row * DIM;
    float s = 0.0f;
    #pragma unroll
    for (int i = 0; i < DIM; i += 32) {
        float v = r[i + lane];
        s += v * v;
    }
    #pragma unroll
    for (int off = 16; off >= 1; off >>= 1) s += __shfl_xor(s, off, 32);
    if (lane == 0) norms[row] = s;
}

// ---------------------------------------------------------------------------
// z = x @ W^T + b   (M=N tokens, 256 cols, K=256 reduction)
// block: 8 waves, wave = 16 rows x 64 cols (4 WMMA tiles), block = 128 rows x 64 cols
// ---------------------------------------------------------------------------
__global__ __launch_bounds__(256) void vq_proj_kernel(
    const float* __restrict__ x, const float* __restrict__ W,
    const float* __restrict__ bias, float* __restrict__ z) {
    const int lane = threadIdx.x & 31;
    const int wave = threadIdx.x >> 5;
    const int hk   = lane >> 4;      // which K-half this lane holds (A/B frag layout)
    const int lm   = lane & 15;      // M (for A) / N (for B) within tile
    const int rbase = blockIdx.x * 128 + wave * 16;
    const int cbase = blockIdx.y * 64;

    v8f acc[4] = {};
    const float* xrow = x + (size_t)(rbase + lm) * DIM;

    for (int k0 = 0; k0 < DIM; k0 += 4) {
        const int ks = k0 + 2 * hk;
        v2f a = *(const v2f*)(xrow + ks);   // A: lane<16 -> K=k0,k0+1 ; lane>=16 -> k0+2,k0+3
        #pragma unroll
        for (int t = 0; t < 4; ++t) {
            const float* wrow = W + (size_t)(cbase + t * 16 + lm) * DIM + ks;
            v2f bf = *(const v2f*)wrow;     // B: same half-split layout, lane = N
            acc[t] = __builtin_amdgcn_wmma_f32_16x16x4_f32(
                false, a, false, bf, (short)0, acc[t], false, false);
        }
    }
    // add bias, store (C/D layout: VGPR j -> M = j + 8*hk, N = lm)
    #pragma unroll
    for (int t = 0; t < 4; ++t) {
        const int c = cbase + t * 16 + lm;
        const float bv = bias[c];
        #pragma unroll
        for (int j = 0; j < 8; ++j) {
            const int r = rbase + j + 8 * hk;
            z[(size_t)r * DIM + c] = acc[t][j] + bv;
        }
    }
}

// ---------------------------------------------------------------------------
// fused distance GEMM + argmin:
//   s = z @ cb^T ; d = ||E||^2 - 2 s  (||z||^2 is per-row constant -> dropped)
// block: 8 waves x 16 rows = 128 rows; each wave scans all K codes in chunks of 64.
// Codebook chunks (64 rows x 1KB = 64KB) are double-buffered through LDS via
// coalesced global_load_async_to_lds_b128; WMMA B-fragments come from LDS with
// an XOR bank swizzle (conflict-free ds_load_b64). A-fragments (the wave's
// 16x256 z tile) are preloaded into registers and reused for all 8192 codes.
// ---------------------------------------------------------------------------
__global__ __launch_bounds__(256, 1) void vq_argmin_kernel(
    const float* __restrict__ z, const float* __restrict__ cb,
    const float* __restrict__ norms, int* __restrict__ idxOut, int K) {
    __shared__ float buf[2][64 * DIM];     // 2 x 64KB double buffer

    const int tid  = threadIdx.x;
    const int lane = tid & 31;
    const int wave = tid >> 5;
    const int hk   = lane >> 4;
    const int lm   = lane & 15;
    const int rbase = blockIdx.x * 128 + wave * 16;

    // Preload all 64 A fragments (K=256 / 4) for this wave's z tile: 128 VGPRs.
    const float* zrow = z + (size_t)(rbase + lm) * DIM;
    v2f afr[64];
    #pragma unroll
    for (int i = 0; i < 64; ++i)
        afr[i] = *(const v2f*)(zrow + 4 * i + 2 * hk);

    float bestv[8];
    int   besti[8];
    #pragma unroll
    for (int j = 0; j < 8; ++j) { bestv[j] = 3.4e38f; besti[j] = 0; }

    // Stage one 64-code chunk: 16 async b128 loads per thread, coalesced over
    // global; LDS dest word = r*256 + 4*(c4 ^ (r&15))  (XOR bank swizzle).
    auto issue_stage = [&](int bsel, int cc) {
        unsigned ldsBase = (unsigned)(uintptr_t)(&buf[bsel][0]);
        #pragma unroll
        for (int s = 0; s < 16; ++s) {
            const int f  = tid + 256 * s;      // 0..4095 b128-groups
            const int r  = f >> 6;             // chunk row 0..63
            const int c4 = f & 63;             // 4-float group in row
            const float* g = cb + (size_t)(cc + r) * DIM + 4 * c4;
            const unsigned lds = ldsBase + 4u * (unsigned)(r * DIM + 4 * (c4 ^ (r & 15)));
            async_load_lds_b128(lds, g);
        }
    };

    issue_stage(0, 0);
    for (int cc = 0; cc < K; cc += 64) {
        const int cur = (cc >> 6) & 1;
        wait_asynccnt0();       // this chunk's LDS data landed (my 16 loads)
        __syncthreads();        // everyone's loads landed; buf[cur^1] fully consumed
        if (cc + 64 < K) issue_stage(cur ^ 1, cc + 64);   // overlap next chunk

        const float* lbuf = &buf[cur][0];
        v8f acc[4] = {};
        #pragma unroll
        for (int i = 0; i < 64; ++i) {
            const int col = (i ^ lm);          // de-swizzle: word 4*(i^lm)+2*hk
            #pragma unroll
            for (int t = 0; t < 4; ++t) {
                const int r = t * 16 + lm;
                v2f bf = *(const v2f*)(lbuf + r * DIM + 4 * col + 2 * hk);
                acc[t] = __builtin_amdgcn_wmma_f32_16x16x4_f32(
                    false, afr[i], false, bf, (short)0, acc[t], false, false);
            }
        }
        #pragma unroll
        for (int t = 0; t < 4; ++t) {
            const int c = cc + t * 16 + lm;
            const float nv = norms[c];
            #pragma unroll
            for (int j = 0; j < 8; ++j) {
                const float d = nv - 2.0f * acc[t][j];   // ties: strict < keeps lowest c
                if (d < bestv[j]) { bestv[j] = d; besti[j] = c; }
            }
        }
    }

    // Reduce the min across the 16 lanes of each half (xor<=8 stays within a half).
    #pragma unroll
    for (int j = 0; j < 8; ++j) {
        float v = bestv[j]; int ix = besti[j];
        #pragma unroll
        for (int off = 8; off >= 1; off >>= 1) {
            float ov = __shfl_xor(v, off, 32);
            int   oi = __shfl_xor(ix, off, 32);
            if (ov < v || (ov == v && oi < ix)) { v = ov; ix = oi; }
        }
        if (lm == 0) idxOut[rbase + j + 8 * hk] = ix;
    }
}

// ---------------------------------------------------------------------------
// gather q = cb[idx], x_recon = z + (q - z), loss partial sums
// ---------------------------------------------------------------------------
__global__ __launch_bounds__(256) void vq_recon_kernel(
    const float* __restrict__ z, const float* __restrict__ cb,
    const int* __restrict__ idx, float* __restrict__ out,
    float* __restrict__ lossAcc) {
    const size_t e = (size_t)blockIdx.x * 256 + threadIdx.x;
    const int row = (int)(e >> 8);
    const int d   = (int)(e & 255);
    const int c   = idx[row];
    const float zv = z[e];
    const float qv = cb[(size_t)c * DIM + d];
    out[e] = zv + (qv - zv);                    // straight-through forward value
    const float diff = zv - qv;
    float s = diff * diff;

    #pragma unroll
    for (int off = 16; off >= 1; off >>= 1) s += __shfl_xor(s, off, 32);
    __shared__ float wsum[8];
    const int lane = threadIdx.x & 31, wave = threadIdx.x >> 5;
    if (lane == 0) wsum[wave] = s;
    __syncthreads();
    if (wave == 0) {
        float t = (lane < 8) ? wsum[lane] : 0.0f;
        #pragma unroll
        for (int off = 4; off >= 1; off >>= 1) t += __shfl_xor(t, off, 32);
        if (lane == 0) atomicAdd(lossAcc, t);
    }
}

__global__ void vq_finalize_kernel(const float* __restrict__ lossAcc,
                                   float* __restrict__ out, unsigned long long NT) {
    if (threadIdx.x == 0) {
        const float m = lossAcc[0] / (float)NT;
        out[NT]     = m;  // dictionary_loss
        out[NT + 1] = m;  // commitment_loss (same forward value)
    }
}

// ---------------------------------------------------------------------------
// launch
// ---------------------------------------------------------------------------
extern "C" void kernel_launch(void* const* d_in, const int* in_sizes, int n_in,
                              void* d_out, int out_size, void* d_ws, size_t ws_size,
                              hipStream_t stream) {
    const float* x  = (const float*)d_in[0];   // [8,4096,256]
    const float* W  = (const float*)d_in[1];   // [256,256]
    const float* b  = (const float*)d_in[2];   // [256]
    const float* cb = (const float*)d_in[3];   // [K,256]
    float* out = (float*)d_out;

    const int N = in_sizes[0] / DIM;           // 32768 tokens
    const int K = in_sizes[3] / DIM;           // 8192 codes

    // workspace: z [N*256] | norms [K] | lossAcc [1] | idx [N]   (~32.2 MB)
    float* z       = (float*)d_ws;
    float* norms   = z + (size_t)N * DIM;
    float* lossAcc = norms + K;
    int*   idx     = (int*)(lossAcc + 1);

    vq_init_kernel<<<1, 32, 0, stream>>>(lossAcc);
    vq_norms_kernel<<<(K + 7) / 8, 256, 0, stream>>>(cb, norms, K);
    vq_proj_kernel<<<dim3(N / 128, DIM / 64), 256, 0, stream>>>(x, W, b, z);
    vq_argmin_kernel<<<N / 128, 256, 0, stream>>>(z, cb, norms, idx, K);
    vq_recon_kernel<<<(unsigned)(((size_t)N * DIM) / 256), 256, 0, stream>>>(z, cb, idx, out, lossAcc);
    vq_finalize_kernel<<<1, 32, 0, stream>>>(lossAcc, out, (unsigned long long)N * DIM);
}